// C_REN_36386962931992
// MI455X (gfx1250) — compile-verified
//
#include <hip/hip_runtime.h>

#define B_TOTAL 65536
#define N_IN 32
#define N_OUT 16
#define N_ST 128
#define N_NL 64
#define EPS_F 1e-3f

typedef float v2f __attribute__((ext_vector_type(2)));
typedef float v8f __attribute__((ext_vector_type(8)));

// workspace layout (float offsets)
#define OFF_H    0           // 320*320 = 102400
#define OFF_INVE 102400      // 128*128 = 16384
#define OFF_G    118784      // 128*224 = 28672
#define OFF_WM   147456      // 56*9*64 = 32256  (swizzled main weights)
#define OFF_WSV  179712      // 56*4*64 = 14336  (swizzled sv weights)
#define OFF_D11  194048      // 64*64   = 4096   (columns of D11, strict-lower)
#define OFF_IL   198144      // 64               (1/Lambda)
// total = 198208 floats = 792832 bytes

// main-kernel dynamic LDS partition (floats)
#define LDS_D11  0                    // 4096
#define LDS_IL   4096                 // 64
#define LDS_W    4160                 // 8 waves * 2 tiles * 16*68 = 17408
#define LDS_TOT  (4160 + 17408)       // 21568 floats = 86272 bytes

__device__ __forceinline__ v8f wmma4(v2f a, v2f b, v8f c) {
  return __builtin_amdgcn_wmma_f32_16x16x4_f32(false, a, false, b, (short)0, c,
                                               false, false);
}

__device__ __forceinline__ float tanh_fast(float x) {
#if defined(__has_builtin)
#if __has_builtin(__builtin_amdgcn_tanhf)
  return __builtin_amdgcn_tanhf(x);
#else
  float e = __expf(2.0f * x);
  return 1.0f - 2.0f / (e + 1.0f);
#endif
#else
  float e = __expf(2.0f * x);
  return 1.0f - 2.0f / (e + 1.0f);
#endif
}

// -------- kernel 1: H = X^T X + eps*I  (320x320) --------
__global__ void __launch_bounds__(256) k_gram(const float* __restrict__ X,
                                              float* __restrict__ H) {
  int idx = blockIdx.x * 256 + threadIdx.x;  // 0..102399
  int a = idx / 320, b = idx % 320;
  float s = 0.0f;
  for (int k = 0; k < 320; ++k) s = fmaf(X[k * 320 + a], X[k * 320 + b], s);
  if (a == b) s += EPS_F;
  H[idx] = s;
}

// -------- kernel 2: invert E = 0.5*(H11+H33+Y1-Y1^T), Gauss-Jordan in LDS ----
__global__ void __launch_bounds__(1024) k_inv(const float* __restrict__ H,
                                              const float* __restrict__ Y1,
                                              float* __restrict__ invE) {
  extern __shared__ float sm[];  // 128 x 258 (augmented [E|I], padded)
  int tid = threadIdx.x;
  for (int idx = tid; idx < 128 * 256; idx += 1024) {
    int r = idx >> 8, c = idx & 255;
    float v;
    if (c < 128)
      v = 0.5f * (H[r * 320 + c] + H[(192 + r) * 320 + 192 + c] +
                  Y1[r * 128 + c] - Y1[c * 128 + r]);
    else
      v = (c - 128 == r) ? 1.0f : 0.0f;
    sm[r * 258 + c] = v;
  }
  __syncthreads();
  int er = tid >> 3;
  int ec = (tid & 7) << 5;
  for (int k = 0; k < 128; ++k) {
    float pinv = 1.0f / sm[k * 258 + k];
    __syncthreads();
    if (tid < 256) sm[k * 258 + tid] *= pinv;
    __syncthreads();
    float f = sm[er * 258 + k];
    __syncthreads();
    if (er != k) {
#pragma unroll 8
      for (int c = ec; c < ec + 32; ++c)
        sm[er * 258 + c] -= f * sm[k * 258 + c];
    }
    __syncthreads();
  }
  for (int idx = tid; idx < 128 * 128; idx += 1024) {
    int r = idx >> 7, c = idx & 127;
    invE[idx] = sm[r * 258 + 128 + c];
  }
}

// -------- kernel 3: G = invE @ [Fm | B1 | B2]  (128 x 224) --------
__global__ void __launch_bounds__(256) k_fuse(const float* __restrict__ H,
                                              const float* __restrict__ invE,
                                              const float* __restrict__ B2,
                                              float* __restrict__ G) {
  int idx = blockIdx.x * 256 + threadIdx.x;  // < 28672
  int n = idx / 224, k = idx % 224;
  float s = 0.0f;
  if (k < 192) {  // Fm[c][k]=H[192+c][k], B1[c][k-128]=H[192+c][k]
    for (int c = 0; c < 128; ++c)
      s = fmaf(invE[n * 128 + c], H[(192 + c) * 320 + k], s);
  } else {
    int k3 = k - 192;
    for (int c = 0; c < 128; ++c)
      s = fmaf(invE[n * 128 + c], B2[c * 32 + k3], s);
  }
  G[idx] = s;
}

// -------- kernel 4: pack weights into WMMA-B-swizzled layout + D11/invLam ----
// swizzle block (kt,nt): 64 floats, value pos = lane*2+c:
//   W[kt*4 + 2*(lane/16) + c][nt*16 + lane%16]
__global__ void __launch_bounds__(256) k_pack(
    const float* __restrict__ H, const float* __restrict__ G,
    const float* __restrict__ C2, const float* __restrict__ D21,
    const float* __restrict__ D22, const float* __restrict__ D12,
    float* __restrict__ Wm, float* __restrict__ Wsv, float* __restrict__ D11c,
    float* __restrict__ IL) {
  int idx = blockIdx.x * 256 + threadIdx.x;
  if (idx < 32256) {  // Wmain: 224 x 144 -> 56 kt x 9 nt
    int blk = idx >> 6, pos = idx & 63;
    int kt = blk / 9, nt = blk % 9;
    int l2 = pos >> 1, c = pos & 1;
    int k = kt * 4 + ((l2 >> 4) << 1) + c;
    int n = nt * 16 + (l2 & 15);
    float v;
    if (n < 128) {
      v = G[n * 224 + k];
    } else {
      int rw = n - 128;
      v = (k < 128)   ? C2[rw * 128 + k]
          : (k < 192) ? D21[rw * 64 + (k - 128)]
                      : D22[rw * 32 + (k - 192)];
    }
    Wm[idx] = v;
  } else if (idx < 46592) {  // Wsv: 224 x 64 -> 56 kt x 4 nt (middle band zero)
    int j = idx - 32256;
    int blk = j >> 6, pos = j & 63;
    int kt = blk >> 2, nt = blk & 3;
    int l2 = pos >> 1, c = pos & 1;
    int k = kt * 4 + ((l2 >> 4) << 1) + c;
    int n = nt * 16 + (l2 & 15);
    float v = (k < 128)   ? -H[(128 + n) * 320 + k]      // C1 = -H21
              : (k < 192) ? 0.0f
                          : D12[n * 32 + (k - 192)];
    Wsv[j] = v;
  } else if (idx < 50688) {  // D11 columns: D11c[i*64+j] = D11[j][i] (j>i)
    int j = idx - 46592;
    int i = j >> 6, jj = j & 63;
    D11c[j] = (jj > i) ? -H[(128 + jj) * 320 + 128 + i] : 0.0f;
  } else if (idx < 50752) {  // invLam = 2 / H22[i][i]
    int i = idx - 50688;
    IL[i] = 2.0f / H[(128 + i) * 320 + 128 + i];
  }
}

// -------- kernel 5: main batched REN step, 2 row-tiles (32 rows) per wave ----
// waves_per_eu(2): cap VGPRs at half the file -> no s_set_vgpr_msb overhead,
// 2 waves/SIMD so recursion VALU of one wave overlaps WMMA of the other.
__global__ void __launch_bounds__(256)
__attribute__((amdgpu_waves_per_eu(2))) k_main(
    const float* __restrict__ state, const float* __restrict__ inpt,
    const float* __restrict__ Wm, const float* __restrict__ Wsv,
    const float* __restrict__ D11c, const float* __restrict__ IL,
    float* __restrict__ out) {
  extern __shared__ float lds[];
  float* sD11 = lds + LDS_D11;
  float* sIL = lds + LDS_IL;
  float* sW = lds + LDS_W;

  int tid = threadIdx.x;
  for (int idx = tid; idx < 4096; idx += 256) sD11[idx] = D11c[idx];
  if (tid < 64) sIL[tid] = IL[tid];
  __syncthreads();

  int wave = tid >> 5, lane = tid & 31;
  int lh = lane >> 4, ln = lane & 15;
  int rowb0 = (blockIdx.x * 8 + wave) * 32;
  int rowb1 = rowb0 + 16;
  const float* s0 = state + (size_t)(rowb0 + ln) * 128 + 2 * lh;
  const float* s1 = state + (size_t)(rowb1 + ln) * 128 + 2 * lh;
  const float* i0 = inpt + (size_t)(rowb0 + ln) * 32 + 2 * lh;
  const float* i1 = inpt + (size_t)(rowb1 + ln) * 32 + 2 * lh;
  const v8f vzero = {};

  // ---- phase 1: vacc = sv = [state|inpt] @ Wsv^T  (D-layout accumulators)
  v8f va0[4], va1[4];
#pragma unroll
  for (int nt = 0; nt < 4; ++nt) {
    va0[nt] = vzero;
    va1[nt] = vzero;
  }
#pragma unroll
  for (int kk = 0; kk < 40; ++kk) {
    const int kt = (kk < 32) ? kk : kk + 16;  // skip zero w-band
    v2f b[4];
#pragma unroll
    for (int nt = 0; nt < 4; ++nt)
      b[nt] = *(const v2f*)(Wsv + (((kt << 2) + nt) << 6) + lane * 2);
    v2f a0, a1;
    if (kk < 32) {
      a0 = *(const v2f*)(s0 + kt * 4);
      a1 = *(const v2f*)(s1 + kt * 4);
    } else {
      a0 = *(const v2f*)(i0 + (kt - 48) * 4);
      a1 = *(const v2f*)(i1 + (kt - 48) * 4);
    }
#pragma unroll
    for (int nt = 0; nt < 4; ++nt) {
      va0[nt] = wmma4(a0, b[nt], va0[nt]);
      va1[nt] = wmma4(a1, b[nt], va1[nt]);
    }
  }

  // ---- phase 2: sequential tanh recursion via ds_bpermute broadcast
  int lmask = lane & 16;
#pragma unroll
  for (int t = 0; t < 4; ++t) {
    for (int ii = 0; ii < 16; ++ii) {  // rolled: reg-constant indexing only
      int i = t * 16 + ii;
      int bidx = (ii + lmask) << 2;
      float il = sIL[i];
      float d0 = sD11[i * 64 + ln];
      float d1 = sD11[i * 64 + 16 + ln];
      float d2 = sD11[i * 64 + 32 + ln];
      float d3 = sD11[i * 64 + 48 + ln];
      float wv0[8], wv1[8];
#pragma unroll
      for (int r = 0; r < 8; ++r) {
        int r0 = __builtin_amdgcn_ds_bpermute(bidx, __float_as_int(va0[t][r]));
        int r1 = __builtin_amdgcn_ds_bpermute(bidx, __float_as_int(va1[t][r]));
        wv0[r] = tanh_fast(__int_as_float(r0) * il);
        wv1[r] = tanh_fast(__int_as_float(r1) * il);
      }
#pragma unroll
      for (int r = 0; r < 8; ++r) {  // D11 cols zero for j<=i -> mask-free
        va0[0][r] += wv0[r] * d0;
        va0[1][r] += wv0[r] * d1;
        va0[2][r] += wv0[r] * d2;
        va0[3][r] += wv0[r] * d3;
        va1[0][r] += wv1[r] * d0;
        va1[1][r] += wv1[r] * d1;
        va1[2][r] += wv1[r] * d2;
        va1[3][r] += wv1[r] * d3;
      }
    }
  }

  // ---- phase 3: w = tanh(v/Lam), stage to LDS in A-layout source form
  float* wl0 = sW + wave * 2176;
  float* wl1 = wl0 + 1088;
#pragma unroll
  for (int t = 0; t < 4; ++t) {
    float il = sIL[t * 16 + ln];
#pragma unroll
    for (int r = 0; r < 8; ++r) {
      wl0[(r + 8 * lh) * 68 + t * 16 + ln] = tanh_fast(va0[t][r] * il);
      wl1[(r + 8 * lh) * 68 + t * 16 + ln] = tanh_fast(va1[t][r] * il);
    }
  }

  // ---- phase 4: [state|w|inpt](224) @ Wm^T -> 144 cols, two N-sweeps
  float* outp = out;
  float* sp = out + (size_t)B_TOTAL * N_OUT;
#pragma unroll
  for (int g = 0; g < 2; ++g) {
    const int ntBeg = (g == 0) ? 0 : 5;
    const int ntNum = (g == 0) ? 5 : 4;
    v8f ac0[5], ac1[5];
#pragma unroll
    for (int j = 0; j < 5; ++j) {
      ac0[j] = vzero;
      ac1[j] = vzero;
    }
#pragma unroll
    for (int kt = 0; kt < 56; ++kt) {
      v2f b[5];
#pragma unroll
      for (int j = 0; j < ntNum; ++j)
        b[j] = *(const v2f*)(Wm + ((kt * 9 + ntBeg + j) << 6) + lane * 2);
      v2f a0, a1;
      if (kt < 32) {
        a0 = *(const v2f*)(s0 + kt * 4);
        a1 = *(const v2f*)(s1 + kt * 4);
      } else if (kt < 48) {
        a0 = *(const v2f*)(wl0 + ln * 68 + (kt - 32) * 4 + 2 * lh);
        a1 = *(const v2f*)(wl1 + ln * 68 + (kt - 32) * 4 + 2 * lh);
      } else {
        a0 = *(const v2f*)(i0 + (kt - 48) * 4);
        a1 = *(const v2f*)(i1 + (kt - 48) * 4);
      }
#pragma unroll
      for (int j = 0; j < ntNum; ++j) {
        ac0[j] = wmma4(a0, b[j], ac0[j]);
        ac1[j] = wmma4(a1, b[j], ac1[j]);
      }
    }
    // ---- stores for this N-sweep: output (B,16) then state_plus (B,128)
#pragma unroll
    for (int j = 0; j < ntNum; ++j) {
      const int nt = ntBeg + j;
      if (nt < 8) {
#pragma unroll
        for (int r = 0; r < 8; ++r) {
          sp[(size_t)(rowb0 + r + 8 * lh) * 128 + nt * 16 + ln] = ac0[j][r];
          sp[(size_t)(rowb1 + r + 8 * lh) * 128 + nt * 16 + ln] = ac1[j][r];
        }
      } else {
#pragma unroll
        for (int r = 0; r < 8; ++r) {
          outp[(size_t)(rowb0 + r + 8 * lh) * 16 + ln] = ac0[j][r];
          outp[(size_t)(rowb1 + r + 8 * lh) * 16 + ln] = ac1[j][r];
        }
      }
    }
  }
}

extern "C" void kernel_launch(void* const* d_in, const int* in_sizes, int n_in,
                              void* d_out, int out_size, void* d_ws,
                              size_t ws_size, hipStream_t stream) {
  (void)in_sizes; (void)n_in; (void)out_size; (void)ws_size;
  const float* inpt = (const float*)d_in[0];
  const float* state = (const float*)d_in[1];
  const float* X = (const float*)d_in[2];
  const float* Y1 = (const float*)d_in[3];
  const float* B2 = (const float*)d_in[4];
  const float* C2 = (const float*)d_in[5];
  const float* D21 = (const float*)d_in[6];
  const float* D22 = (const float*)d_in[7];
  const float* D12 = (const float*)d_in[8];
  float* ws = (float*)d_ws;
  float* H = ws + OFF_H;
  float* invE = ws + OFF_INVE;
  float* G = ws + OFF_G;
  float* Wm = ws + OFF_WM;
  float* Wsv = ws + OFF_WSV;
  float* D11c = ws + OFF_D11;
  float* IL = ws + OFF_IL;

  k_gram<<<400, 256, 0, stream>>>(X, H);
  k_inv<<<1, 1024, 128 * 258 * 4, stream>>>(H, Y1, invE);
  k_fuse<<<112, 256, 0, stream>>>(H, invE, B2, G);
  k_pack<<<199, 256, 0, stream>>>(H, G, C2, D21, D22, D12, Wm, Wsv, D11c, IL);
  k_main<<<256, 256, LDS_TOT * 4, stream>>>(state, inpt, Wm, Wsv, D11c, IL,
                                            (float*)d_out);
}